// AttentionPool_18665927868749
// MI455X (gfx1250) — compile-verified
//
#include <hip/hip_runtime.h>
#include <math.h>

// AttentionPool, B=32 S=512 N=64 D=64.
// Key algebraic folding: only k = kv[..., :D] is used, and only inside a full
// contraction with q. So  prod[b,s] = ( <x[b,s,:,:], r> + c0 ) / 8  with
//   r = (qe @ Wq^T + bq) @ Wkv[:D,:]   (two 64x64x64 fp32 GEMMs -> WMMA)
//   c0 = sum_{n,d} q[n,d] * bkv[d]
// Then online-softmax single pass over x (256 MB read ONCE, ~11 us at 23.3 TB/s)
// with flash-style partial (m, z, acc) records merged in a final kernel.

typedef __attribute__((ext_vector_type(2))) float v2f;
typedef __attribute__((ext_vector_type(8))) float v8f;

#define S_LEN 512
#define ROW_ELEMS 4096          // N*D
#define REC_STRIDE 4104         // 4096 acc + m + z + pad (16B aligned)

// ---------------------------------------------------------------------------
// Kernel 1: build r (64x64) and c0 using V_WMMA_F32_16X16X4_F32.
// 16 waves; wave w owns output tile (ti=w/4, tj=w%4).
// f32 WMMA layouts (ISA 7.12.2):
//   A 16x4 : lane<16: M=lane, comp0->K=kk+0, comp1->K=kk+1
//            lane>=16: M=lane-16, comp0->K=kk+2, comp1->K=kk+3
//   B 4x16 : lane<16: N=lane, comp0->K=kk+0, comp1->K=kk+1 ; upper half K=kk+2/3
//   C/D    : vgpr j: lanes 0-15 -> (M=j, N=lane); lanes 16-31 -> (M=j+8, N=lane-16)
// ---------------------------------------------------------------------------
__global__ __launch_bounds__(512) void k_build_r(
    const float* __restrict__ qe,    // (64,64)
    const float* __restrict__ Wq_w,  // (64,64)
    const float* __restrict__ Wq_b,  // (64)
    const float* __restrict__ Wkv_w, // (128,64), only rows [0,64) used
    const float* __restrict__ Wkv_b, // (128), only [0,64) used
    float* __restrict__ r_out,       // (64,64) -> ws
    float* __restrict__ c0_out)      // scalar  -> ws
{
    __shared__ float qs[64 * 64];
    __shared__ float red[512];

    const int tid  = threadIdx.x;
    const int wave = tid >> 5;
    const int lane = tid & 31;
    const int half = lane >> 4;      // 0: lanes 0-15, 1: lanes 16-31
    const int lrow = lane & 15;
    const int ti   = wave >> 2;
    const int tj   = wave & 3;
    const int arow = ti * 16 + lrow; // A-matrix row this lane supplies
    const int bcol = tj * 16 + lrow; // B/D column this lane supplies

    // ---- q = qe @ Wq^T : A[m,k] = qe[m,k], B[k,n] = Wq_w[n,k] ----
    v8f c;
    #pragma unroll
    for (int j = 0; j < 8; ++j) c[j] = 0.0f;
    #pragma unroll
    for (int kk = 0; kk < 64; kk += 4) {
        v2f a, b;
        const int k0 = kk + 2 * half;
        a.x = qe[arow * 64 + k0];
        a.y = qe[arow * 64 + k0 + 1];
        b.x = Wq_w[bcol * 64 + k0];          // B[k, bcol] = Wq_w[bcol, k]
        b.y = Wq_w[bcol * 64 + k0 + 1];
        c = __builtin_amdgcn_wmma_f32_16x16x4_f32(false, a, false, b,
                                                  (short)0, c, false, false);
    }
    const float bias = Wq_b[bcol];
    #pragma unroll
    for (int j = 0; j < 8; ++j) {
        const int row = ti * 16 + half * 8 + j;
        qs[row * 64 + bcol] = c[j] + bias;
    }
    __syncthreads();

    // ---- r = q @ Wk : A[m,k] = q[m,k] (LDS), B[k,n] = Wkv_w[k,n] ----
    #pragma unroll
    for (int j = 0; j < 8; ++j) c[j] = 0.0f;
    #pragma unroll
    for (int kk = 0; kk < 64; kk += 4) {
        v2f a, b;
        const int k0 = kk + 2 * half;
        a.x = qs[arow * 64 + k0];
        a.y = qs[arow * 64 + k0 + 1];
        b.x = Wkv_w[(k0) * 64 + bcol];
        b.y = Wkv_w[(k0 + 1) * 64 + bcol];
        c = __builtin_amdgcn_wmma_f32_16x16x4_f32(false, a, false, b,
                                                  (short)0, c, false, false);
    }
    #pragma unroll
    for (int j = 0; j < 8; ++j) {
        const int row = ti * 16 + half * 8 + j;
        r_out[row * 64 + bcol] = c[j];
    }

    // ---- c0 = sum_{n,d} q[n,d] * Wkv_b[d] ----
    float p = 0.0f;
    #pragma unroll
    for (int i = 0; i < 8; ++i) {
        const int idx = tid * 8 + i;             // 512 threads * 8 = 4096
        p += qs[idx] * Wkv_b[idx & 63];
    }
    red[tid] = p;
    __syncthreads();
    if (tid == 0) {
        float s = 0.0f;
        for (int i = 0; i < 512; ++i) s += red[i];
        *c0_out = s;
    }
}

// ---------------------------------------------------------------------------
// Kernel 2: single-pass online softmax + weighted accumulation.
// grid = 32 * nchunk blocks, 256 threads (8 waves). Each block streams
// srows = 512/nchunk rows of x[b]; the 4096-float row lives in registers
// (16 floats/thread as 4 float4, stride-256 for coalescing).
// ---------------------------------------------------------------------------
__global__ __launch_bounds__(256) void k_flash(
    const float* __restrict__ x,
    const float* __restrict__ rvec,
    const float* __restrict__ c0p,
    float* __restrict__ partials,
    int nchunk, int srows)
{
    const int b     = blockIdx.x / nchunk;
    const int chunk = blockIdx.x % nchunk;
    const int tid   = threadIdx.x;
    const int wave  = tid >> 5;
    const int lane  = tid & 31;

    __shared__ float red[8];
    __shared__ float bc[2];

    // r stays in registers for the whole block lifetime
    const float4* r4 = (const float4*)rvec;
    float4 rr[4];
    #pragma unroll
    for (int c = 0; c < 4; ++c) rr[c] = r4[tid + 256 * c];

    float4 acc[4];
    #pragma unroll
    for (int c = 0; c < 4; ++c) acc[c] = make_float4(0.f, 0.f, 0.f, 0.f);

    const float c0 = *c0p;
    float m = -__builtin_inff();
    float z = 0.0f;

    const int s0 = chunk * srows;
    for (int s = s0; s < s0 + srows; ++s) {
        const float4* xr = (const float4*)x + ((size_t)(b * S_LEN + s)) * 1024;
        float4 xv[4];
        #pragma unroll
        for (int c = 0; c < 4; ++c) xv[c] = xr[tid + 256 * c];

        // dot(x_row, r): per-thread partial
        float dot = 0.0f;
        #pragma unroll
        for (int c = 0; c < 4; ++c) {
            dot = fmaf(xv[c].x, rr[c].x, dot);
            dot = fmaf(xv[c].y, rr[c].y, dot);
            dot = fmaf(xv[c].z, rr[c].z, dot);
            dot = fmaf(xv[c].w, rr[c].w, dot);
        }
        // wave32 reduction, then cross-wave via LDS
        #pragma unroll
        for (int off = 16; off >= 1; off >>= 1)
            dot += __shfl_down(dot, off, 32);
        if (lane == 0) red[wave] = dot;
        __syncthreads();
        if (tid == 0) {
            float p = red[0];
            #pragma unroll
            for (int w = 1; w < 8; ++w) p += red[w];
            p = (p + c0) * 0.125f;               // / sqrt(64)
            const float mn = fmaxf(m, p);
            const float sc = __expf(m - mn);     // 0 on first iter (m = -inf)
            const float e  = __expf(p - mn);
            z = z * sc + e;
            m = mn;
            bc[0] = sc;
            bc[1] = e;
        }
        __syncthreads();
        const float sc = bc[0];
        const float e  = bc[1];
        if (sc == 1.0f) {                        // common case: max unchanged
            #pragma unroll
            for (int c = 0; c < 4; ++c) {
                acc[c].x = fmaf(e, xv[c].x, acc[c].x);
                acc[c].y = fmaf(e, xv[c].y, acc[c].y);
                acc[c].z = fmaf(e, xv[c].z, acc[c].z);
                acc[c].w = fmaf(e, xv[c].w, acc[c].w);
            }
        } else {
            #pragma unroll
            for (int c = 0; c < 4; ++c) {
                acc[c].x = fmaf(e, xv[c].x, acc[c].x * sc);
                acc[c].y = fmaf(e, xv[c].y, acc[c].y * sc);
                acc[c].z = fmaf(e, xv[c].z, acc[c].z * sc);
                acc[c].w = fmaf(e, xv[c].w, acc[c].w * sc);
            }
        }
    }

    float* rec = partials + (size_t)(b * nchunk + chunk) * REC_STRIDE;
    float4* rec4 = (float4*)rec;
    #pragma unroll
    for (int c = 0; c < 4; ++c) rec4[tid + 256 * c] = acc[c];
    if (tid == 0) {
        rec[4096] = m;
        rec[4097] = z;
    }
}

// ---------------------------------------------------------------------------
// Kernel 3: merge flash partials per batch, write out[b,n,d].
// ---------------------------------------------------------------------------
__global__ __launch_bounds__(256) void k_combine(
    const float* __restrict__ partials,
    float* __restrict__ out,
    int nchunk)
{
    const int b   = blockIdx.x;
    const int tid = threadIdx.x;

    float M = -__builtin_inff();
    for (int i = 0; i < nchunk; ++i)
        M = fmaxf(M, partials[(size_t)(b * nchunk + i) * REC_STRIDE + 4096]);

    float Z = 0.0f;
    float4 acc[4];
    #pragma unroll
    for (int c = 0; c < 4; ++c) acc[c] = make_float4(0.f, 0.f, 0.f, 0.f);

    for (int i = 0; i < nchunk; ++i) {
        const float* rec = partials + (size_t)(b * nchunk + i) * REC_STRIDE;
        const float w = __expf(rec[4096] - M);
        Z += w * rec[4097];
        const float4* rec4 = (const float4*)rec;
        #pragma unroll
        for (int c = 0; c < 4; ++c) {
            const float4 v = rec4[tid + 256 * c];
            acc[c].x = fmaf(w, v.x, acc[c].x);
            acc[c].y = fmaf(w, v.y, acc[c].y);
            acc[c].z = fmaf(w, v.z, acc[c].z);
            acc[c].w = fmaf(w, v.w, acc[c].w);
        }
    }

    const float inv = 1.0f / Z;
    float4* o4 = (float4*)out + (size_t)b * 1024;
    #pragma unroll
    for (int c = 0; c < 4; ++c) {
        acc[c].x *= inv; acc[c].y *= inv; acc[c].z *= inv; acc[c].w *= inv;
        o4[tid + 256 * c] = acc[c];
    }
}

// ---------------------------------------------------------------------------
extern "C" void kernel_launch(void* const* d_in, const int* in_sizes, int n_in,
                              void* d_out, int out_size, void* d_ws, size_t ws_size,
                              hipStream_t stream) {
    const float* x     = (const float*)d_in[0];  // (32,512,64,64)
    const float* qe    = (const float*)d_in[1];  // (64,64)
    const float* Wq_w  = (const float*)d_in[2];  // (64,64)
    const float* Wq_b  = (const float*)d_in[3];  // (64)
    const float* Wkv_w = (const float*)d_in[4];  // (128,64)
    const float* Wkv_b = (const float*)d_in[5];  // (128)
    float* out = (float*)d_out;                  // (32,64,64)

    float* W        = (float*)d_ws;
    float* r        = W;                         // 4096 floats
    float* c0       = W + 4096;                  // 1 float
    float* partials = W + 8192;                  // 32*nchunk records

    // Pick largest s-split that fits in workspace (16 -> ~8.4 MB).
    int nchunk = 1;
    const int cands[5] = {16, 8, 4, 2, 1};
    for (int i = 0; i < 5; ++i) {
        const size_t need = ((size_t)8192 + (size_t)32 * cands[i] * REC_STRIDE) * 4;
        if (need <= ws_size) { nchunk = cands[i]; break; }
    }

    k_build_r<<<1, 512, 0, stream>>>(qe, Wq_w, Wq_b, Wkv_w, Wkv_b, r, c0);
    k_flash<<<32 * nchunk, 256, 0, stream>>>(x, r, c0, partials, nchunk,
                                             S_LEN / nchunk);
    k_combine<<<32, 256, 0, stream>>>(partials, out, nchunk);
}